// MultiHopGCN_51032801411302
// MI455X (gfx1250) — compile-verified
//
#include <hip/hip_runtime.h>

typedef __attribute__((ext_vector_type(2))) float v2f;
typedef __attribute__((ext_vector_type(8))) float v8f;

#define D 64

// ---------------- degree / normalization ----------------

__global__ __launch_bounds__(256) void k_deg_init(float* deg, int n) {
  int i = blockIdx.x * 256 + threadIdx.x;
  if (i < n) deg[i] = 1.0f;  // self-loop weight contributes 1 to every degree
}

__global__ __launch_bounds__(256) void k_deg_edge(const int* __restrict__ dst,
                                                  const float* __restrict__ w,
                                                  float* __restrict__ deg, int nE) {
  int e = blockIdx.x * 256 + threadIdx.x;
  if (e < nE) atomicAdd(&deg[dst[e]], w[e]);
}

__global__ __launch_bounds__(256) void k_deg_fin(float* dis, int n) {
  int i = blockIdx.x * 256 + threadIdx.x;
  if (i < n) {
    float dg = dis[i];
    dis[i] = dg > 0.f ? rsqrtf(fmaxf(dg, 1e-12f)) : 0.f;
  }
}

// ---------------- WMMA GEMM: t = (relu?)h @ W  [+ fused out = t*dis^2 + b] ----
// Block = 256 threads = 8 waves; each wave computes 16 rows x 64 cols.
// W (64x64 f32, 16KB) staged in LDS once per block.

template <bool RELU, bool FUSE>
__global__ __launch_bounds__(256) void k_gemm_wmma(const float* __restrict__ h,
                                                   const float* __restrict__ W,
                                                   float* __restrict__ t,
                                                   const float* __restrict__ dis,
                                                   const float* __restrict__ bias,
                                                   float* __restrict__ out,
                                                   int nNodes) {
  __shared__ float Wl[D * D];
  {
    const float4* Wv = (const float4*)W;
    float4* Lv = (float4*)Wl;
    for (int i = threadIdx.x; i < (D * D / 4); i += 256) Lv[i] = Wv[i];
  }
  __syncthreads();  // all waves pass the barrier before any wave exits

  const int wave = threadIdx.x >> 5;
  const int lane = threadIdx.x & 31;
  const int rowBase = (blockIdx.x * 8 + wave) * 16;
  if (rowBase >= nNodes) return;  // uniform per wave -> EXEC all-1s inside

  const int hs  = lane >> 4;   // 0: K pair (k,k+1), 1: (k+2,k+3)
  const int l15 = lane & 15;
  const float* __restrict__ hrow = h + (size_t)(rowBase + l15) * D + (hs << 1);

  v8f c0 = {}, c1 = {}, c2 = {}, c3 = {};
#pragma unroll
  for (int k4 = 0; k4 < 16; ++k4) {
    const int k = k4 * 4;
    float ax = hrow[k];
    float ay = hrow[k + 1];
    if (RELU) { ax = fmaxf(ax, 0.f); ay = fmaxf(ay, 0.f); }
    v2f a; a[0] = ax; a[1] = ay;

    const int krow = k + (hs << 1);
    const float* wb = &Wl[krow * D + l15];
    v2f b0, b1, b2, b3;
    b0[0] = wb[0];  b0[1] = wb[D];
    b1[0] = wb[16]; b1[1] = wb[D + 16];
    b2[0] = wb[32]; b2[1] = wb[D + 32];
    b3[0] = wb[48]; b3[1] = wb[D + 48];

    c0 = __builtin_amdgcn_wmma_f32_16x16x4_f32(false, a, false, b0, (short)0, c0, false, false);
    c1 = __builtin_amdgcn_wmma_f32_16x16x4_f32(false, a, false, b1, (short)0, c1, false, false);
    c2 = __builtin_amdgcn_wmma_f32_16x16x4_f32(false, a, false, b2, (short)0, c2, false, false);
    c3 = __builtin_amdgcn_wmma_f32_16x16x4_f32(false, a, false, b3, (short)0, c3, false, false);
  }

  // C/D layout: VGPR i -> row rowBase + i + 8*hs, col = 16*j + l15
  float bi0, bi1, bi2, bi3;
  if (FUSE) {
    bi0 = bias[l15];
    bi1 = bias[16 + l15];
    bi2 = bias[32 + l15];
    bi3 = bias[48 + l15];
  }
#pragma unroll
  for (int i = 0; i < 8; ++i) {
    const int row = rowBase + i + (hs << 3);
    const size_t r = (size_t)row * D;
    t[r + l15]      = c0[i];
    t[r + 16 + l15] = c1[i];
    t[r + 32 + l15] = c2[i];
    t[r + 48 + l15] = c3[i];
    if (FUSE) {
      const float di = dis[row];
      const float d2 = di * di;
      out[r + l15]      = fmaf(c0[i], d2, bi0);
      out[r + 16 + l15] = fmaf(c1[i], d2, bi1);
      out[r + 32 + l15] = fmaf(c2[i], d2, bi2);
      out[r + 48 + l15] = fmaf(c3[i], d2, bi3);
    }
  }
}

// ---------------- self-loop + bias init (layer 3 only): out = t*dis^2 + b ----

__global__ __launch_bounds__(256) void k_self_bias(const float* __restrict__ t,
                                                   const float* __restrict__ dis,
                                                   const float* __restrict__ bias,
                                                   float* __restrict__ out, int total) {
  int i = blockIdx.x * 256 + threadIdx.x;
  if (i < total) {
    float di = dis[i >> 6];
    out[i] = fmaf(t[i], di * di, bias[i & 63]);
  }
}

// ---------------- edge scatter: 16 lanes/edge, float4 per lane ----------------

__global__ __launch_bounds__(256) void k_edge_scatter(const int* __restrict__ src,
                                                      const int* __restrict__ dst,
                                                      const float* __restrict__ w,
                                                      const float* __restrict__ dis,
                                                      const float* __restrict__ t,
                                                      float* __restrict__ out, int nE) {
  const long long gid = (long long)blockIdx.x * 256 + threadIdx.x;
  const int e = (int)(gid >> 4);
  if (e >= nE) return;
  const int c4 = ((int)gid & 15) << 2;
  const int s = src[e];
  const int d = dst[e];
  const float coef = dis[s] * w[e] * dis[d];
  const float4 v = *(const float4*)(t + (size_t)s * D + c4);
  float* op = out + (size_t)d * D + c4;
  atomicAdd(op,     v.x * coef);
  atomicAdd(op + 1, v.y * coef);
  atomicAdd(op + 2, v.z * coef);
  atomicAdd(op + 3, v.w * coef);
}

// ---------------- launch ----------------

extern "C" void kernel_launch(void* const* d_in, const int* in_sizes, int n_in,
                              void* d_out, int out_size, void* d_ws, size_t ws_size,
                              hipStream_t stream) {
  const float* x  = (const float*)d_in[0];
  const int*   ei = (const int*)d_in[1];   // [2, nE]: src then dst
  const float* ew = (const float*)d_in[2];
  const float* W1 = (const float*)d_in[3];
  const float* b1 = (const float*)d_in[4];
  const float* W2 = (const float*)d_in[5];
  const float* b2 = (const float*)d_in[6];
  const float* W3 = (const float*)d_in[7];
  const float* b3 = (const float*)d_in[8];

  const int nN = in_sizes[0] / D;
  const int nE = in_sizes[2];
  const int* src = ei;
  const int* dst = ei + nE;

  float* dis  = (float*)d_ws;
  float* tbuf = dis + (size_t)((nN + 63) & ~63);
  float* hbuf = tbuf + (size_t)nN * D;
  float* out  = (float*)d_out;

  const int gNode  = (nN + 255) / 256;
  const int gEdge  = (nE + 255) / 256;
  const int total  = nN * D;
  const int gTotal = (total + 255) / 256;
  const int gGemm  = (nN + 127) / 128;
  const int gScat  = (int)(((long long)nE * 16 + 255) / 256);

  // normalization: dis = D^{-1/2}
  k_deg_init<<<gNode, 256, 0, stream>>>(dis, nN);
  k_deg_edge<<<gEdge, 256, 0, stream>>>(dst, ew, dis, nE);
  k_deg_fin <<<gNode, 256, 0, stream>>>(dis, nN);

  // layer 1: x -> hbuf   (GEMM + fused self-loop/bias init of hbuf)
  k_gemm_wmma<false, true><<<gGemm, 256, 0, stream>>>(x, W1, tbuf, dis, b1, hbuf, nN);
  k_edge_scatter<<<gScat, 256, 0, stream>>>(src, dst, ew, dis, tbuf, hbuf, nE);

  // layer 2: relu(hbuf) -> d_out (h2 storage; fused init, no aliasing)
  k_gemm_wmma<true, true><<<gGemm, 256, 0, stream>>>(hbuf, W2, tbuf, dis, b2, out, nN);
  k_edge_scatter<<<gScat, 256, 0, stream>>>(src, dst, ew, dis, tbuf, out, nE);

  // layer 3: relu(h2=d_out) -> d_out. Unfused: GEMM consumes h2 fully, then
  // the separate init pass overwrites d_out (no restrict aliasing in-kernel).
  k_gemm_wmma<true, false><<<gGemm, 256, 0, stream>>>(out, W3, tbuf, dis, b3, nullptr, nN);
  k_self_bias<<<gTotal, 256, 0, stream>>>(tbuf, dis, b3, out, total);
  k_edge_scatter<<<gScat, 256, 0, stream>>>(src, dst, ew, dis, tbuf, out, nE);
}